// S4Model_19808389169783
// MI455X (gfx1250) — compile-verified
//
#include <hip/hip_runtime.h>
#include <hip/hip_bf16.h>
#include <math.h>

#define L_MAX 4096
#define BATCH 32
#define D_IN  128
#define H_DIM 256
#define N_ST  64

typedef __attribute__((ext_vector_type(2))) float v2f;
typedef __attribute__((ext_vector_type(8))) float v8f;

#if defined(__gfx1250__) && \
    __has_builtin(__builtin_amdgcn_global_load_async_to_lds_b32) && \
    __has_builtin(__builtin_amdgcn_s_wait_asynccnt)
#define S4_ASYNC 1
typedef __attribute__((address_space(1))) int g_int;   // global
typedef __attribute__((address_space(3))) int l_int;   // LDS
#else
#define S4_ASYNC 0
#endif

// ---------------------------------------------------------------------------
// Kernel 1: discretize SSM parameters per (h,n).
// lam = exp(dt*A),  C_disc = C*(lam-1)/A.  Store [lam_r, lam_i, 2*Cd_r, 2*Cd_i].
// ---------------------------------------------------------------------------
__global__ void s4_params_kernel(const float* __restrict__ log_dt,
                                 const float* __restrict__ A_log_real,
                                 const float* __restrict__ A_imag,
                                 const float* __restrict__ C_real,
                                 const float* __restrict__ C_imag,
                                 float4* __restrict__ params) {
    int i = blockIdx.x * blockDim.x + threadIdx.x;   // i = h*N + n
    if (i >= H_DIM * N_ST) return;
    int h = i / N_ST;
    float dt   = __expf(log_dt[h]);
    float Ar   = -__expf(A_log_real[i]);
    float Aim  = A_imag[i];
    float dtAr = Ar * dt, dtAi = Aim * dt;
    float er = __expf(dtAr);
    float lr = er * __cosf(dtAi);
    float li = er * __sinf(dtAi);
    // (lam - 1) / A   via conj(A)/|A|^2
    float nr  = lr - 1.0f, ni = li;
    float inv = 1.0f / (Ar * Ar + Aim * Aim);
    float qr  = (nr * Ar + ni * Aim) * inv;
    float qi  = (ni * Ar - nr * Aim) * inv;
    float cr = C_real[i], ci = C_imag[i];
    float cdr = cr * qr - ci * qi;
    float cdi = cr * qi + ci * qr;
    params[i] = make_float4(lr, li, 2.0f * cdr, 2.0f * cdi);
}

// ---------------------------------------------------------------------------
// Kernel 2: encoder GEMM via V_WMMA_F32_16X16X4_F32 (native fp32 tensor op).
// Computes u transposed:  ut[b][h][l] = sum_d W[d][h] * x[l][b][d]
// Tile: M = 16 h-rows (A = W^T), N = 16 l-cols (B = x slice), K = 128 (32 WMMAs).
// ---------------------------------------------------------------------------
__global__ __launch_bounds__(256)
void s4_encode_wmma(const float* __restrict__ x,
                    const float* __restrict__ W,
                    float* __restrict__ ut) {
    const int b    = blockIdx.x;        // 0..31
    const int h0   = blockIdx.y * 16;   // h tile base
    const int wave = threadIdx.x >> 5;
    const int lane = threadIdx.x & 31;
    const int m     = lane & 15;        // A row (h) == B col (l)
    const int khalf = (lane >> 4) * 2;  // 0 or 2

    for (int lt = wave; lt < L_MAX / 16; lt += 8) {
        const int l0 = lt * 16;
        v8f acc = {};
        #pragma unroll 4
        for (int k0 = 0; k0 < D_IN; k0 += 4) {
            const int k = k0 + khalf;
            v2f a, bm;
            a.x = W[(size_t)(k    ) * H_DIM + h0 + m];
            a.y = W[(size_t)(k + 1) * H_DIM + h0 + m];
            const float* xp = x + ((size_t)(l0 + m) * BATCH + b) * D_IN + k;
            bm.x = xp[0];
            bm.y = xp[1];
            acc = __builtin_amdgcn_wmma_f32_16x16x4_f32(
                false, a, false, bm, (short)0, acc, false, false);
        }
        const int row_h = h0 + (lane >> 4) * 8;
        float* op = ut + ((size_t)b * H_DIM + row_h) * (size_t)L_MAX + l0 + m;
        #pragma unroll
        for (int r = 0; r < 8; ++r)
            op[(size_t)r * L_MAX] = acc[r];
    }
}

// ---------------------------------------------------------------------------
// Kernel 3: diagonal SSM scan + skip + gelu.  One wave per (b,h); lane owns
// states n=lane and n=lane+32.  u is double-buffered into LDS with
// GLOBAL_LOAD_ASYNC_TO_LDS_B32 (ASYNCcnt-tracked), per-step scalar broadcast
// comes from a conflict-free LDS read; per-step cross-lane xor-reduction for y.
// ---------------------------------------------------------------------------
__global__ __launch_bounds__(256)
void s4_scan_kernel(const float* __restrict__ ut,
                    const float4* __restrict__ params,
                    const float* __restrict__ D_skip,
                    float* __restrict__ out) {
    __shared__ float lds_u[8 * 64];          // 8 waves x 2 buffers x 32 floats
    const int wave = threadIdx.x >> 5;
    const int lane = threadIdx.x & 31;
    const int gid  = blockIdx.x * 8 + wave;  // (b,h) pair
    const int b = gid >> 8;                  // / H_DIM
    const int h = gid & (H_DIM - 1);

    const float4 p0 = params[h * N_ST + lane];
    const float4 p1 = params[h * N_ST + lane + 32];
    const float dsk = D_skip[h];
    const float* up = ut + ((size_t)b * H_DIM + h) * (size_t)L_MAX;
    float* lbuf = &lds_u[wave * 64];

    float s0r = 0.f, s0i = 0.f, s1r = 0.f, s1i = 0.f;

#if S4_ASYNC
    // prologue: stage chunk 0
    __builtin_amdgcn_global_load_async_to_lds_b32(
        (g_int*)(up + lane), (l_int*)(lbuf + lane), 0, 0);
#endif

    for (int c = 0; c < L_MAX / 32; ++c) {
        const int l0  = c * 32;
        const int cur = (c & 1) * 32;

#if S4_ASYNC
        if (c + 1 < L_MAX / 32) {
            // prefetch next chunk into the other half-buffer, keep 1 in flight
            __builtin_amdgcn_global_load_async_to_lds_b32(
                (g_int*)(up + l0 + 32 + lane),
                (l_int*)(lbuf + (((c + 1) & 1) * 32) + lane), 0, 0);
            __builtin_amdgcn_s_wait_asynccnt(1);
        } else {
            __builtin_amdgcn_s_wait_asynccnt(0);
        }
        asm volatile("" ::: "memory");
#else
        if (l0 + 64 <= L_MAX)
            __builtin_prefetch(up + l0 + 32 + lane, 0, 1);  // global_prefetch_b8
        const float uv = up[l0 + lane];
#endif

        float ybuf = 0.f;
        #pragma unroll 4
        for (int t = 0; t < 32; ++t) {
#if S4_ASYNC
            const float u = lbuf[cur + t];          // LDS broadcast read
#else
            const float u = __shfl(uv, t, 32);
#endif
            float nr0 = fmaf(p0.x, s0r, fmaf(-p0.y, s0i, u));
            float ni0 = fmaf(p0.x, s0i, p0.y * s0r);
            s0r = nr0; s0i = ni0;
            float nr1 = fmaf(p1.x, s1r, fmaf(-p1.y, s1i, u));
            float ni1 = fmaf(p1.x, s1i, p1.y * s1r);
            s1r = nr1; s1i = ni1;
            float part = fmaf(p0.z, s0r,
                         fmaf(-p0.w, s0i,
                         fmaf(p1.z, s1r, -p1.w * s1i)));
            part += __shfl_xor(part, 1, 32);
            part += __shfl_xor(part, 2, 32);
            part += __shfl_xor(part, 4, 32);
            part += __shfl_xor(part, 8, 32);
            part += __shfl_xor(part, 16, 32);
            const float y = fmaf(dsk, u, part);
            // gelu (tanh approximation, matches jax.nn.gelu default)
            const float inner = 0.7978845608028654f * fmaf(0.044715f, y * y * y, y);
            const float g = 0.5f * y * (1.0f + tanhf(inner));
            if (lane == t) ybuf = g;
        }
        // out shape (L, B, H): out[l][b][h]
        out[((size_t)(l0 + lane) * BATCH + b) * H_DIM + h] = ybuf;
    }
}

// ---------------------------------------------------------------------------
extern "C" void kernel_launch(void* const* d_in, const int* in_sizes, int n_in,
                              void* d_out, int out_size, void* d_ws, size_t ws_size,
                              hipStream_t stream) {
    const float* x      = (const float*)d_in[0];
    const float* W_enc  = (const float*)d_in[1];
    const float* log_dt = (const float*)d_in[2];
    const float* Alr    = (const float*)d_in[3];
    const float* Aim    = (const float*)d_in[4];
    const float* Cr     = (const float*)d_in[5];
    const float* Ci     = (const float*)d_in[6];
    const float* Dsk    = (const float*)d_in[7];
    float* out = (float*)d_out;

    // workspace layout: [params: H*N float4 = 256KB][ut: B*H*L f32 = 128MB]
    float4* params = (float4*)d_ws;
    float*  ut     = (float*)((char*)d_ws + (size_t)H_DIM * N_ST * sizeof(float4));

    s4_params_kernel<<<(H_DIM * N_ST + 255) / 256, 256, 0, stream>>>(
        log_dt, Alr, Aim, Cr, Ci, params);

    dim3 g2(BATCH, H_DIM / 16);
    s4_encode_wmma<<<g2, 256, 0, stream>>>(x, W_enc, ut);

    s4_scan_kernel<<<(BATCH * H_DIM) / 8, 256, 0, stream>>>(ut, params, Dsk, out);
}